// CycleEnergyDirectAddNoHead_80590766342750
// MI455X (gfx1250) — compile-verified
//
#include <hip/hip_runtime.h>
#include <cstdint>
#include <cstddef>

// Problem constants (match reference)
constexpr int   kP    = 8;
constexpr int   kU    = 256;
constexpr int   kV    = 256;
constexpr int   kF    = 256 * 49;   // 12544
constexpr float kCOEF = 0.1f;

typedef __attribute__((ext_vector_type(2))) float v2f;
typedef __attribute__((ext_vector_type(8))) float v8f;

// ---------------------------------------------------------------------------
// Kernel 1: vv[p*V + v] = sum_f feat_v[p,v,f]^2   (one 256-thread block / row)
// ---------------------------------------------------------------------------
__global__ void vv_kernel(const float* __restrict__ fv, float* __restrict__ vv) {
  const int row = blockIdx.x;                       // 0 .. P*V-1
  const float* p = fv + (size_t)row * kF;
  float s = 0.f;
  for (int i = threadIdx.x; i < kF; i += 256) {
    const float t = p[i];
    s = fmaf(t, t, s);
  }
  // wave32 butterfly reduce
  for (int off = 16; off > 0; off >>= 1)
    s += __shfl_xor(s, off, 32);
  __shared__ float partial[8];
  if ((threadIdx.x & 31) == 0) partial[threadIdx.x >> 5] = s;
  __syncthreads();
  if (threadIdx.x == 0) {
    float t = 0.f;
    #pragma unroll
    for (int w = 0; w < 8; ++w) t += partial[w];
    vv[row] = t;
  }
}

// ---------------------------------------------------------------------------
// Kernel 2: GEMM1 (uv = U x Vt, K = F) fused with dist epilogue.
// One wave32 per 16x16 output tile. A/B fragments are contiguous float2 loads.
// dist = COEF * (vv[v] - 2*uv)   (uu term cancels in the softmax that follows)
// ---------------------------------------------------------------------------
__global__ __launch_bounds__(32)
void gemm1_dist(const float* __restrict__ fu, const float* __restrict__ fv,
                const float* __restrict__ vv, float* __restrict__ sc) {
  const int p    = blockIdx.z;
  const int u0   = blockIdx.y << 4;
  const int v0   = blockIdx.x << 4;
  const int lane = threadIdx.x;          // 0..31, full wave
  const int r    = lane & 15;            // A: row M, B: col N
  const int kk   = (lane >> 4) << 1;     // contiguous K-pair base: 0 or 2

  const float* Au = fu + ((size_t)p * kU + u0 + r) * kF + kk;
  const float* Bv = fv + ((size_t)p * kV + v0 + r) * kF + kk;  // B = feat_v^T

  v8f acc = {};
  #pragma unroll 4
  for (int k = 0; k < kF; k += 4) {
    const v2f a = *reinterpret_cast<const v2f*>(Au + k);
    const v2f b = *reinterpret_cast<const v2f*>(Bv + k);
    acc = __builtin_amdgcn_wmma_f32_16x16x4_f32(false, a, false, b,
                                                (short)0, acc, false, false);
  }

  // C/D layout: VGPR i -> M = i + 8*(lane>=16), N = lane&15
  const float vvn   = vv[p * kV + v0 + r];
  const int   mbase = (lane >> 4) << 3;
  float* out = sc + ((size_t)p * kU + u0 + mbase) * kV + v0 + r;
  #pragma unroll
  for (int i = 0; i < 8; ++i)
    out[(size_t)i * kV] = kCOEF * (vvn - 2.0f * acc[i]);
}

// ---------------------------------------------------------------------------
// Kernel 3: in-place softmax along V (row length 256). One wave per row,
// 8 elements per lane, butterfly max/sum reductions.
// ---------------------------------------------------------------------------
__global__ void softmax_v(float* __restrict__ sc) {
  const int row  = (blockIdx.x << 3) + (threadIdx.x >> 5);  // 0 .. P*U-1
  const int lane = threadIdx.x & 31;
  float* rp = sc + (size_t)row * kV;

  float x[8];
  float m = -3.402823466e38f;
  #pragma unroll
  for (int i = 0; i < 8; ++i) { x[i] = rp[lane + (i << 5)]; m = fmaxf(m, x[i]); }
  for (int off = 16; off > 0; off >>= 1) m = fmaxf(m, __shfl_xor(m, off, 32));

  float s = 0.f;
  #pragma unroll
  for (int i = 0; i < 8; ++i) { x[i] = __expf(x[i] - m); s += x[i]; }
  for (int off = 16; off > 0; off >>= 1) s += __shfl_xor(s, off, 32);

  const float inv = 1.0f / s;
  #pragma unroll
  for (int i = 0; i < 8; ++i) rp[lane + (i << 5)] = x[i] * inv;
}

// ---------------------------------------------------------------------------
// Kernel 4: GEMM2  neg_fea = score (UxV) @ feat_v (VxF).  K = V = 256.
// One wave32 per 16x16 output tile; A is contiguous float2, B is F-strided.
// Output is streamed with non-temporal stores (write-once, ~103 MB) so it does
// not evict the heavily reused feat_v working set from the 192 MB L2.
// ---------------------------------------------------------------------------
__global__ __launch_bounds__(32)
void gemm2_agg(const float* __restrict__ sc, const float* __restrict__ fv,
               float* __restrict__ outneg) {
  const int p    = blockIdx.z;
  const int u0   = blockIdx.y << 4;
  const int n0   = blockIdx.x << 4;
  const int lane = threadIdx.x;
  const int r    = lane & 15;
  const int kk   = (lane >> 4) << 1;

  const float* As = sc + ((size_t)p * kU + u0 + r) * kV + kk;
  const float* Bv = fv + ((size_t)p * kV + kk) * kF + n0 + r;

  v8f acc = {};
  #pragma unroll 4
  for (int k = 0; k < kV; k += 4) {
    const v2f a = *reinterpret_cast<const v2f*>(As + k);
    v2f b;
    b.x = Bv[(size_t)k * kF];
    b.y = Bv[(size_t)k * kF + kF];
    acc = __builtin_amdgcn_wmma_f32_16x16x4_f32(false, a, false, b,
                                                (short)0, acc, false, false);
  }

  const int mbase = (lane >> 4) << 3;
  float* out = outneg + ((size_t)p * kU + u0 + mbase) * kF + n0 + r;
  #pragma unroll
  for (int i = 0; i < 8; ++i)
    __builtin_nontemporal_store(acc[i], out + (size_t)i * kF);
}

// ---------------------------------------------------------------------------
extern "C" void kernel_launch(void* const* d_in, const int* in_sizes, int n_in,
                              void* d_out, int out_size, void* d_ws, size_t ws_size,
                              hipStream_t stream) {
  const float* feat_u = (const float*)d_in[0];
  const float* feat_v = (const float*)d_in[1];
  float* out = (float*)d_out;

  // workspace: vv (P*V floats) | scores (P*U*V floats)  => ~2.01 MB
  float* vv = (float*)d_ws;
  float* sc = vv + kP * kV;

  // pos_fea half of the output is a straight copy of feat_u
  hipMemcpyAsync(out, feat_u, (size_t)kP * kU * kF * sizeof(float),
                 hipMemcpyDeviceToDevice, stream);

  vv_kernel<<<kP * kV, 256, 0, stream>>>(feat_v, vv);

  dim3 g1(kV / 16, kU / 16, kP);               // 16 x 16 x 8 tiles
  gemm1_dist<<<g1, 32, 0, stream>>>(feat_u, feat_v, vv, sc);

  softmax_v<<<(kP * kU) / 8, 256, 0, stream>>>(sc);

  dim3 g2(kF / 16, kU / 16, kP);               // 784 x 16 x 8 tiles
  gemm2_agg<<<g2, 32, 0, stream>>>(sc, feat_v, out + (size_t)kP * kU * kF);
}